// EdgeMaskDecoder_83769042141419
// MI455X (gfx1250) — compile-verified
//
#include <hip/hip_runtime.h>
#include <hip/hip_bf16.h>
#include <math.h>

typedef __attribute__((ext_vector_type(2))) float v2f;
typedef __attribute__((ext_vector_type(8))) float v8f;

#define HID 128          // hidden dim (fixed by reference)
#define MT_PER_BLOCK 8   // 8 waves/block, one 16-row M-tile per wave

// ---------------------------------------------------------------------------
// Kernel 1: Z = H @ W + b   via V_WMMA_F32_16X16X4_F32 (exact f32 path)
//   grid.x covers M-tiles/8, grid.y in {0,1} selects (Ws,bs,zs)/(Wt,bt,zt)
// ---------------------------------------------------------------------------
__global__ void __launch_bounds__(256)
gemm_bias_wmma(const float* __restrict__ hid,
               const float* __restrict__ Ws, const float* __restrict__ bs,
               const float* __restrict__ Wt, const float* __restrict__ bt,
               float* __restrict__ zs, float* __restrict__ zt,
               int mtiles)
{
    const float* W = (blockIdx.y == 0) ? Ws : Wt;
    const float* b = (blockIdx.y == 0) ? bs : bt;
    float*       Z = (blockIdx.y == 0) ? zs : zt;

    // Stage the full 128x128 weight matrix in LDS (64 KB of the 320 KB/WGP).
    __shared__ float4 ldsW4[HID * HID / 4];
    const float4* Wv = (const float4*)W;
    for (int i = threadIdx.x; i < HID * HID / 4; i += 256)
        ldsW4[i] = Wv[i];
    __syncthreads();
    const float* ldsW = (const float*)ldsW4;

    const int lane = threadIdx.x & 31;
    const int wv   = threadIdx.x >> 5;
    const int mt   = blockIdx.x * MT_PER_BLOCK + wv;   // wave-uniform
    if (mt >= mtiles) return;                          // whole-wave skip: EXEC stays all-1s

    const int l16  = lane & 15;
    const int koff = (lane < 16) ? 0 : 2;              // A/B K-lane split (ISA 7.12.2)
    const int row  = mt * 16 + l16;                    // A-matrix row for this lane
    const float* hrow = hid + (size_t)row * HID;

    v8f acc[8] = {};                                   // 8 N-tiles of 16x16 f32 accum

    for (int k = 0; k < HID; k += 4) {
        // A fragment: lane holds H[row][k+koff], H[row][k+koff+1] -> one b64 load
        float2 a2 = *(const float2*)(hrow + k + koff);
        v2f av; av.x = a2.x; av.y = a2.y;
        const float* wr0 = ldsW + (k + koff) * HID;    // W row k+koff
        const float* wr1 = wr0 + HID;                  // W row k+koff+1
#pragma unroll
        for (int n = 0; n < 8; ++n) {
            const int col = n * 16 + l16;
            v2f bv; bv.x = wr0[col]; bv.y = wr1[col];
            acc[n] = __builtin_amdgcn_wmma_f32_16x16x4_f32(
                false, av, false, bv, (short)0, acc[n], false, false);
        }
    }

    // D layout: VGPR r -> row (mt*16 + r + (lane<16 ? 0 : 8)), col = n*16 + l16
    const int rbase = mt * 16 + ((lane < 16) ? 0 : 8);
#pragma unroll
    for (int n = 0; n < 8; ++n) {
        const int col  = n * 16 + l16;
        const float bb = b[col];
#pragma unroll
        for (int r = 0; r < 8; ++r)
            Z[(size_t)(rbase + r) * HID + col] = acc[n][r] + bb;
    }
}

// ---------------------------------------------------------------------------
// Kernel 2: per-edge gather + dot + sigmoid.  One wave32 per edge:
//   each lane loads float4 of zs[src] and zt[dst] (512B coalesced, L2-resident),
//   5-step shuffle reduction, lane 0 writes sigmoid(logit).
// ---------------------------------------------------------------------------
__global__ void __launch_bounds__(256)
edge_dot_sigmoid(const float* __restrict__ zs, const float* __restrict__ zt,
                 const int* __restrict__ ei, float* __restrict__ out, int E)
{
    const int wave = (int)((blockIdx.x * blockDim.x + threadIdx.x) >> 5);
    const int lane = threadIdx.x & 31;
    if (wave >= E) return;

    const int s = ei[wave];       // edge_index[0][e]
    const int t = ei[E + wave];   // edge_index[1][e]

    const float4 a = *(const float4*)(zs + (size_t)s * HID + lane * 4);
    const float4 c = *(const float4*)(zt + (size_t)t * HID + lane * 4);
    float p = a.x * c.x + a.y * c.y + a.z * c.z + a.w * c.w;

#pragma unroll
    for (int off = 16; off > 0; off >>= 1)
        p += __shfl_xor(p, off, 32);

    if (lane == 0)
        out[wave] = 1.0f / (1.0f + __expf(-p));
}

// ---------------------------------------------------------------------------
extern "C" void kernel_launch(void* const* d_in, const int* in_sizes, int n_in,
                              void* d_out, int out_size, void* d_ws, size_t ws_size,
                              hipStream_t stream)
{
    const float* hid = (const float*)d_in[0];
    const int*   ei  = (const int*)d_in[1];
    const float* Ws  = (const float*)d_in[2];
    const float* bs  = (const float*)d_in[3];
    const float* Wt  = (const float*)d_in[4];
    const float* bt  = (const float*)d_in[5];
    float*       out = (float*)d_out;

    const int N = in_sizes[0] / HID;   // 100000 nodes
    const int E = in_sizes[1] / 2;     // 1.6M edges

    // Scratch: zs then zt, 51.2 MB each -> both fit the 192 MB L2 together,
    // making the edge-gather phase L2-bandwidth bound instead of HBM bound.
    float* zs = (float*)d_ws;
    float* zt = zs + (size_t)N * HID;

    const int mtiles = (N + 15) / 16;                       // 6250
    const int gx     = (mtiles + MT_PER_BLOCK - 1) / MT_PER_BLOCK;
    gemm_bias_wmma<<<dim3(gx, 2), 256, 0, stream>>>(hid, Ws, bs, Wt, bt, zs, zt, mtiles);

    const int eblocks = (E + 7) / 8;                        // 8 waves/block
    edge_dot_sigmoid<<<eblocks, 256, 0, stream>>>(zs, zt, ei, out, E);
}